// OrdinalDoseLoss_43851616092489
// MI455X (gfx1250) — compile-verified
//
#include <hip/hip_runtime.h>
#include <hip/hip_bf16.h>

// Problem constants (match reference).
#define NROWS 262144
#define DEMB  512
#define NCOMP 1024
#define NDOSE 8
#define NSEG  (NCOMP * NDOSE)
#define MARGIN 0.1f

typedef float v2f __attribute__((ext_vector_type(2)));
typedef float v8f __attribute__((ext_vector_type(8)));

// ---------------------------------------------------------------------------
// Pass 1: column sums of E (for mu). 128 threads/block, each thread owns 4
// consecutive columns (float4), blocks grid-stride over rows. Fully coalesced
// 2 KB row reads; 512 atomic f32 adds per block at the end.
// ---------------------------------------------------------------------------
__global__ void __launch_bounds__(128)
colsum_kernel(const float* __restrict__ E, float* __restrict__ colsum) {
    const int c4 = threadIdx.x;            // columns [4*c4 .. 4*c4+3]
    float ax = 0.f, ay = 0.f, az = 0.f, aw = 0.f;
    for (int row = blockIdx.x; row < NROWS; row += gridDim.x) {
        const float4 v = *reinterpret_cast<const float4*>(
            E + (size_t)row * DEMB + (size_t)c4 * 4);
        ax += v.x; ay += v.y; az += v.z; aw += v.w;
    }
    atomicAdd(&colsum[c4 * 4 + 0], ax);
    atomicAdd(&colsum[c4 * 4 + 1], ay);
    atomicAdd(&colsum[c4 * 4 + 2], az);
    atomicAdd(&colsum[c4 * 4 + 3], aw);
}

// ---------------------------------------------------------------------------
// Pass 1b: origin = (colsum/N) / max(||colsum/N||, 1e-12). One 512-thread block.
// ---------------------------------------------------------------------------
__global__ void __launch_bounds__(512)
origin_kernel(const float* __restrict__ colsum, float* __restrict__ origin) {
    __shared__ float red[512];
    const int t = threadIdx.x;
    const float mu = colsum[t] * (1.0f / (float)NROWS);
    red[t] = mu * mu;
    __syncthreads();
    for (int s = 256; s > 0; s >>= 1) {
        if (t < s) red[t] += red[t + s];
        __syncthreads();
    }
    const float norm = fmaxf(sqrtf(red[0]), 1e-12f);
    origin[t] = mu / norm;
}

// ---------------------------------------------------------------------------
// Pass 2: per-row dist = 1 - e_i . origin via V_WMMA_F32_16X16X4_F32.
// Each wave owns a 16-row tile. A-matrix per ISA layout: lane L holds
// A[M = L%16][K = 2*(L/16) + {0,1}] in its v2f. B-matrix: origin slice
// broadcast to all 16 columns, so D[M][N] = dot(row M, origin) for every N;
// lanes 0 and 16 (N==0) hold rows 0-7 / 8-15 in their v8f accumulator and do
// the segment scatter with global f32 atomics.
// ---------------------------------------------------------------------------
__global__ void __launch_bounds__(256)
dist_seg_kernel(const float* __restrict__ E,
                const int*   __restrict__ comp,
                const int*   __restrict__ dose,
                const float* __restrict__ origin,
                float* __restrict__ seg_sum,
                float* __restrict__ seg_cnt) {
    __shared__ float s_origin[DEMB];
    for (int i = threadIdx.x; i < DEMB; i += blockDim.x) s_origin[i] = origin[i];
    __syncthreads();

    const int lane       = threadIdx.x & 31;
    const int waveInBlk  = threadIdx.x >> 5;
    const int wavesPerBlk = blockDim.x >> 5;
    const int waveGlobal = blockIdx.x * wavesPerBlk + waveInBlk;
    const int totalWaves = gridDim.x * wavesPerBlk;
    const int nTiles     = NROWS / 16;

    const int mrow = lane & 15;        // tile row this lane supplies for A
    const int koff = (lane >> 4) * 2;  // K sub-offset: 0 (lanes 0-15) or 2

    for (int tile = waveGlobal; tile < nTiles; tile += totalWaves) {
        const int row0 = tile * 16;
        const float* aptr = E + (size_t)(row0 + mrow) * DEMB + koff;

        v8f acc = {0.f, 0.f, 0.f, 0.f, 0.f, 0.f, 0.f, 0.f};
#pragma unroll 8
        for (int k = 0; k < DEMB; k += 4) {
            const v2f a = *reinterpret_cast<const v2f*>(aptr + k);
            v2f b;
            b.x = s_origin[k + koff];
            b.y = s_origin[k + koff + 1];
            // 8 args: (neg_a, A, neg_b, B, c_mod, C, reuse_a, reuse_b)
            acc = __builtin_amdgcn_wmma_f32_16x16x4_f32(
                false, a, false, b, (short)0, acc, false, false);
        }

        if ((lane & 15) == 0) {        // lanes 0 and 16 hold column N==0 of D
            const int rbase = row0 + (lane >> 4) * 8;
#pragma unroll
            for (int m = 0; m < 8; ++m) {
                const int r = rbase + m;
                const int s = comp[r] * NDOSE + dose[r];
                atomicAdd(&seg_sum[s], 1.0f - acc[m]);
                atomicAdd(&seg_cnt[s], 1.0f);
            }
        }
    }
}

// ---------------------------------------------------------------------------
// Pass 3: ordinal margin scan. One 1024-thread block; thread c = compound c
// walks its 8 doses exactly like the reference loop, then block-reduce.
// ---------------------------------------------------------------------------
__global__ void __launch_bounds__(1024)
ordinal_loss_kernel(const float* __restrict__ seg_sum,
                    const float* __restrict__ seg_cnt,
                    float* __restrict__ out) {
    __shared__ float sl[1024];
    __shared__ float sc[1024];
    const int c = threadIdx.x;

    float last_mean = 0.f, loss = 0.f, cntv = 0.f;
    bool  last_valid = false;
#pragma unroll
    for (int d = 0; d < NDOSE; ++d) {
        const int   s    = c * NDOSE + d;
        const float n    = seg_cnt[s];
        const bool  pres = (n > 0.f);
        const float cur  = seg_sum[s] / fmaxf(n, 1.0f);
        const bool  valid = pres && last_valid;
        const float viol  = fmaxf(MARGIN - (cur - last_mean), 0.0f);
        if (valid) { loss += viol; cntv += 1.0f; }
        if (pres) last_mean = cur;
        last_valid = last_valid || pres;
    }
    sl[c] = loss;
    sc[c] = cntv;
    __syncthreads();
    for (int s2 = 512; s2 > 0; s2 >>= 1) {
        if (c < s2) { sl[c] += sl[c + s2]; sc[c] += sc[c + s2]; }
        __syncthreads();
    }
    if (c == 0) {
        const float L = sl[0], K = sc[0];
        out[0] = (K > 0.f) ? (L / fmaxf(K, 1.0f)) : 0.0f;
    }
}

// ---------------------------------------------------------------------------
// Launcher. Workspace layout (floats): [0,512) colsum | [512,1024) origin |
// [1024,9216) seg_sum | [9216,17408) seg_cnt. Zeroed every call (graph-safe
// hipMemsetAsync) so replays are deterministic in work performed.
// ---------------------------------------------------------------------------
extern "C" void kernel_launch(void* const* d_in, const int* in_sizes, int n_in,
                              void* d_out, int out_size, void* d_ws, size_t ws_size,
                              hipStream_t stream) {
    (void)in_sizes; (void)n_in; (void)out_size; (void)ws_size;

    const float* E    = (const float*)d_in[0];
    const int*   comp = (const int*)d_in[1];
    const int*   dose = (const int*)d_in[2];

    float* ws      = (float*)d_ws;
    float* colsum  = ws;            // 512
    float* origin  = ws + 512;      // 512
    float* seg_sum = ws + 1024;     // 8192
    float* seg_cnt = ws + 9216;     // 8192

    hipMemsetAsync(d_ws, 0, (size_t)(512 + 512 + NSEG + NSEG) * sizeof(float), stream);

    colsum_kernel<<<1024, 128, 0, stream>>>(E, colsum);
    origin_kernel<<<1, 512, 0, stream>>>(colsum, origin);
    dist_seg_kernel<<<512, 256, 0, stream>>>(E, comp, dose, origin, seg_sum, seg_cnt);
    ordinal_loss_kernel<<<1, 1024, 0, stream>>>(seg_sum, seg_cnt, (float*)d_out);
}